// AttentionalPropagation_71648644432027
// MI455X (gfx1250) — compile-verified
//
#include <hip/hip_runtime.h>

// ---------------------------------------------------------------------------
// AttentionalPropagation for MI455X (gfx1250, wave32, WMMA).
// GEMMs: V_WMMA_F32_16X16X4_F32 (full f32 — problem is tiny/latency bound).
// B panels staged to LDS via GLOBAL_LOAD_ASYNC_TO_LDS_B128 (ASYNCcnt path)
// when the toolchain exposes the builtin; vectorized sync fill otherwise.
// All matrices have leading dimension 512 (compile-time) so address math
// folds into global_load immediate offsets.
// ---------------------------------------------------------------------------

typedef __attribute__((ext_vector_type(2))) float v2f;
typedef __attribute__((ext_vector_type(8))) float v8f;
typedef __attribute__((ext_vector_type(4))) int   v4i;

#define LDS_AS __attribute__((address_space(3)))

#if defined(__gfx1250__) &&                                            \
    __has_builtin(__builtin_amdgcn_global_load_async_to_lds_b128) &&   \
    __has_builtin(__builtin_amdgcn_s_wait_asynccnt)
#define HAVE_ASYNC_LDS 1
#else
#define HAVE_ASYNC_LDS 0
#endif

#define DD 256
#define NQ 512
#define NK 512
#define TD 512   // 2*DD
#define LD 512   // common leading dimension of every matrix in this problem

// workspace layout (float offsets)
#define WS_AQ    0
#define WS_AK    (WS_AQ + DD * NQ)        // 131072
#define WS_PROB  (WS_AK + DD * NK)        // 262144
#define WS_CAT   (WS_PROB + NQ * NK)      // 524288
#define WS_H2    (WS_CAT + TD * NQ)       // 786432
// total = 1048576 floats = 4 MiB

// LDS panel layouts (single raw array, 16B-aligned rows for b128 stores):
//   BTRANS=0: element(kk,c) at sraw[kk*20 + c]   (64 rows x 20, pad 4)
//   BTRANS=1: element(kk,c) at sraw[c*68 + kk]   (16 rows x 68, pad 4)
#define SB_ROW  20
#define SBT_ROW 68

// ---------------------------------------------------------------------------
// Wave-tile f32 WMMA GEMM:  C[m,n] = act(sum_k A[m,k]*B[k,n] + bias[m])
//   A: row-major ld=512.  B: BTRANS=0 -> B[k*512+n]; BTRANS=1 -> B[n*512+k].
// Block = 128 threads = 4 waves; all waves share n0, each owns one 16-row
// m-tile => B panel identical across waves: stage a 64-deep K panel in LDS
// once per block (async b128 memory->LDS), feed WMMA B-fragments from LDS.
// A-fragment (16x4 f32): lane L<16 -> row L, K=k..k+1; lanes 16..31 ->
// row L-16, K=k+2..k+3.  C/D: VGPR v -> row m0+(L/16)*8+v, col n0+(L%16).
// ---------------------------------------------------------------------------
template <int BTRANS, int RELU, int BIAS, int K>
__global__ __launch_bounds__(128) void gemm_wmma_f32(
    const float* __restrict__ A, const float* __restrict__ B,
    const float* __restrict__ bias, float* __restrict__ C) {
  __shared__ __align__(16) float sraw[64 * SB_ROW];  // 5 KB, fits both layouts

  const int tid  = threadIdx.x;
  const int lane = tid & 31;
  const int wave = tid >> 5;
  const int half = lane >> 4;  // 0 or 1
  const int l16  = lane & 15;

  const int n0 = blockIdx.x * 16;
  const int m0 = (blockIdx.y * 4 + wave) * 16;

  const float* Arow = A + (size_t)(m0 + l16) * LD;
  v8f acc = {};

#pragma unroll 1
  for (int kp = 0; kp < K; kp += 64) {
    __syncthreads();  // previous panel fully consumed

    // ---- stage 64x16 B panel: 256 b128 quads, 2 per thread ----
#if HAVE_ASYNC_LDS
    if (BTRANS) {
#pragma unroll
      for (int j = 0; j < 2; ++j) {
        const int qq = tid + j * 128;
        const int c = qq >> 4, kq = (qq & 15) << 2;  // 4 consecutive k
        __builtin_amdgcn_global_load_async_to_lds_b128(
            (v4i*)(B + (size_t)(n0 + c) * LD + kp + kq),
            (v4i LDS_AS*)&sraw[c * SBT_ROW + kq], 0, 0);
      }
    } else {
#pragma unroll
      for (int j = 0; j < 2; ++j) {
        const int qq = tid + j * 128;
        const int kk = qq >> 2, cq = (qq & 3) << 2;  // 4 consecutive n
        __builtin_amdgcn_global_load_async_to_lds_b128(
            (v4i*)(B + (size_t)(kp + kk) * LD + n0 + cq),
            (v4i LDS_AS*)&sraw[kk * SB_ROW + cq], 0, 0);
      }
    }
    __builtin_amdgcn_s_wait_asynccnt(0);
#else
    if (BTRANS) {
#pragma unroll
      for (int j = 0; j < 2; ++j) {
        const int qq = tid + j * 128;
        const int c = qq >> 4, kq = (qq & 15) << 2;
        const float4 v = *(const float4*)(B + (size_t)(n0 + c) * LD + kp + kq);
        *(float4*)&sraw[c * SBT_ROW + kq] = v;
      }
    } else {
#pragma unroll
      for (int j = 0; j < 2; ++j) {
        const int qq = tid + j * 128;
        const int kk = qq >> 2, cq = (qq & 3) << 2;
        const float4 v = *(const float4*)(B + (size_t)(kp + kk) * LD + n0 + cq);
        *(float4*)&sraw[kk * SB_ROW + cq] = v;
      }
    }
#endif
    __syncthreads();

    // speculative prefetch of the next A panel (dropped if past the end)
    __builtin_prefetch(Arow + kp + 64, 0, 1);

#pragma unroll
    for (int kk = 0; kk < 64; kk += 4) {
      v2f a, b;
      const float* ap = Arow + kp + kk + half * 2;
      a.x = ap[0];
      a.y = ap[1];
      if (BTRANS) {  // adjacent in LDS -> ds_load_b64
        b.x = sraw[l16 * SBT_ROW + kk + half * 2];
        b.y = sraw[l16 * SBT_ROW + kk + half * 2 + 1];
      } else {       // 16 lanes hit 16 distinct banks (row pad 20)
        b.x = sraw[(kk + half * 2) * SB_ROW + l16];
        b.y = sraw[(kk + half * 2 + 1) * SB_ROW + l16];
      }
      acc = __builtin_amdgcn_wmma_f32_16x16x4_f32(false, a, false, b, (short)0,
                                                  acc, false, false);
    }
  }

  float* ccol = C + n0 + l16;
#pragma unroll
  for (int v = 0; v < 8; ++v) {
    const int m = m0 + half * 8 + v;
    float val = acc[v];
    if (BIAS) val += bias[m];
    if (RELU) val = fmaxf(val, 0.0f);
    ccol[(size_t)m * LD] = val;
  }
}

// ---------------------------------------------------------------------------
// scores[n,m] = b2 + sum_d w2[d] * relu(aq[d,n] + ak[d,m])
// (relu inside the reduction -> not a GEMM; VALU with LDS-staged panels)
// Block: 256 threads = 16(n) x 16(m) tile; stage 16-wide columns of aq/ak.
// ---------------------------------------------------------------------------
__global__ __launch_bounds__(256) void scores_kernel(
    const float* __restrict__ aq, const float* __restrict__ ak,
    const float* __restrict__ w2, const float* __restrict__ b2,
    float* __restrict__ scores) {
  __shared__ float sAQ[DD][16];  // 16 KB
  __shared__ float sAK[DD][16];  // 16 KB
  __shared__ float sW2[DD];

  const int tid = threadIdx.x;
  const int nl  = tid & 15;   // local n
  const int ml  = tid >> 4;   // local m
  const int n0  = blockIdx.x * 16;
  const int m0  = blockIdx.y * 16;

  for (int i = tid; i < DD * 16; i += 256) {
    const int d = i >> 4, c = i & 15;
    sAQ[d][c] = aq[d * NQ + n0 + c];
    sAK[d][c] = ak[d * NK + m0 + c];
  }
  for (int i = tid; i < DD; i += 256) sW2[i] = w2[i];
  __syncthreads();

  float acc = 0.0f;
#pragma unroll 8
  for (int d = 0; d < DD; ++d) {
    const float h = sAQ[d][nl] + sAK[d][ml];
    acc = fmaf(sW2[d], fmaxf(h, 0.0f), acc);
  }
  scores[(size_t)(n0 + nl) * NK + m0 + ml] = acc + b2[0];
}

// ---------------------------------------------------------------------------
// Row softmax over m (NK=512). One block (256 threads) per row n.
// Wave32 shfl reductions + tiny LDS cross-wave combine.
// ---------------------------------------------------------------------------
__global__ __launch_bounds__(256) void softmax_kernel(
    const float* __restrict__ scores, float* __restrict__ prob) {
  __shared__ float red[8];
  const int n   = blockIdx.x;
  const int tid = threadIdx.x;
  const float* row = scores + (size_t)n * NK;

  const float a = row[tid];
  const float b = row[tid + 256];

  float m = fmaxf(a, b);
#pragma unroll
  for (int off = 16; off > 0; off >>= 1) m = fmaxf(m, __shfl_xor(m, off, 32));
  if ((tid & 31) == 0) red[tid >> 5] = m;
  __syncthreads();
  float mfull = red[0];
#pragma unroll
  for (int i = 1; i < 8; ++i) mfull = fmaxf(mfull, red[i]);
  __syncthreads();

  const float ea = __expf(a - mfull);
  const float eb = __expf(b - mfull);
  float s = ea + eb;
#pragma unroll
  for (int off = 16; off > 0; off >>= 1) s += __shfl_xor(s, off, 32);
  if ((tid & 31) == 0) red[tid >> 5] = s;
  __syncthreads();
  float sfull = 0.0f;
#pragma unroll
  for (int i = 0; i < 8; ++i) sfull += red[i];
  const float inv = 1.0f / sfull;

  prob[(size_t)n * NK + tid]       = ea * inv;
  prob[(size_t)n * NK + tid + 256] = eb * inv;
}

// ---------------------------------------------------------------------------
extern "C" void kernel_launch(void* const* d_in, const int* in_sizes, int n_in,
                              void* d_out, int out_size, void* d_ws,
                              size_t ws_size, hipStream_t stream) {
  const float* x      = (const float*)d_in[0];  // (1, 256, 512)
  const float* source = (const float*)d_in[1];  // (1, 256, 512)
  const float* w1     = (const float*)d_in[2];  // (256, 512)
  const float* b1     = (const float*)d_in[3];  // (256,)
  const float* w2     = (const float*)d_in[4];  // (1, 256)
  const float* b2     = (const float*)d_in[5];  // (1,)
  const float* wa     = (const float*)d_in[6];  // (512, 512)
  const float* ba     = (const float*)d_in[7];  // (512,)
  const float* wb     = (const float*)d_in[8];  // (256, 512)
  const float* bb     = (const float*)d_in[9];  // (256,)

  float* out_y      = (float*)d_out;      // 256*512
  float* out_scores = out_y + DD * NQ;    // 512*512

  float* ws   = (float*)d_ws;
  float* AQ   = ws + WS_AQ;
  float* AK   = ws + WS_AK;
  float* PROB = ws + WS_PROB;
  float* CAT  = ws + WS_CAT;
  float* H2   = ws + WS_H2;

  // 1) aq = w1[:, :256] @ x + b1   (M=256, N=512, K=256)
  gemm_wmma_f32<0, 0, 1, DD><<<dim3(NQ / 16, DD / 64), 128, 0, stream>>>(
      w1, x, b1, AQ);

  // 2) ak = w1[:, 256:] @ source   (M=256, N=512, K=256)
  gemm_wmma_f32<0, 0, 0, DD><<<dim3(NK / 16, DD / 64), 128, 0, stream>>>(
      w1 + DD, source, nullptr, AK);

  // 3) scores (written directly to the output slot; softmax reads it back)
  scores_kernel<<<dim3(NQ / 16, NK / 16), 256, 0, stream>>>(AQ, AK, w2, b2,
                                                            out_scores);

  // 4) prob = softmax(scores, axis=-1)
  softmax_kernel<<<NQ, 256, 0, stream>>>(out_scores, PROB);

  // 5) cat[0:256] = x (device-to-device, stream-ordered)
  (void)hipMemcpyAsync(CAT, x, (size_t)DD * NQ * sizeof(float),
                       hipMemcpyDeviceToDevice, stream);

  // 6) cat[256:512] = message = source @ prob^T  (M=256, N=512, K=512, B^T)
  gemm_wmma_f32<1, 0, 0, NK><<<dim3(NQ / 16, DD / 64), 128, 0, stream>>>(
      source, PROB, nullptr, CAT + DD * NQ);

  // 7) h2 = relu(wa @ cat + ba)    (M=512, N=512, K=512)
  gemm_wmma_f32<0, 1, 1, TD><<<dim3(NQ / 16, TD / 64), 128, 0, stream>>>(
      wa, CAT, ba, H2);

  // 8) y = wb @ h2 + bb            (M=256, N=512, K=512)
  gemm_wmma_f32<0, 0, 1, TD><<<dim3(NQ / 16, DD / 64), 128, 0, stream>>>(
      wb, H2, bb, out_y);
}